// ReconciliationBridge_8521215115945
// MI455X (gfx1250) — compile-verified
//
#include <hip/hip_runtime.h>
#include <hip/hip_bf16.h>
#include <cstdint>
#include <cstddef>

typedef float v2f __attribute__((ext_vector_type(2)));
typedef float v8f __attribute__((ext_vector_type(8)));

#define D        128
#define NNODES   100000
#define NEDGES   1000000
#define KE       384
#define KN       256
#define LN_EPS   1e-5f
#define CNT_EPS  1e-10f

// LDS stage: 64-row x 128-col chunk of W, stored transposed sB[col*SB_STRIDE + k]
// stride 66 floats: keeps 8-byte alignment for float2 reads and avoids bank conflicts
#define SB_STRIDE 66

__device__ __forceinline__ float half16_reduce(float x) {
  // butterfly sum across the 16 lanes of this half-wave (wave32)
  x += __shfl_xor(x, 1, 32);
  x += __shfl_xor(x, 2, 32);
  x += __shfl_xor(x, 4, 32);
  x += __shfl_xor(x, 8, 32);
  return x;
}

__global__ void zero_kernel(float* __restrict__ p, size_t n) {
  size_t i = (size_t)blockIdx.x * blockDim.x + threadIdx.x;
  size_t stride = (size_t)gridDim.x * blockDim.x;
  for (; i < n; i += stride) p[i] = 0.0f;
}

// ---------------- Edge update: new_e = LN(e + [e, x_src, x_tgt] @ We + be) ---------
// plus atomic scatter of new_e into eSum/eCnt for both endpoints.
__global__ __launch_bounds__(256) void edge_kernel(
    const float* __restrict__ nodeF, const float* __restrict__ edgeF,
    const int* __restrict__ srcI, const int* __restrict__ tgtI,
    const float* __restrict__ We, const float* __restrict__ be,
    const float* __restrict__ gammaE, const float* __restrict__ betaE,
    float* __restrict__ newE, float* __restrict__ eSum, float* __restrict__ eCnt)
{
  __shared__ float sB[D * SB_STRIDE];
  const int tid  = threadIdx.x;
  const int lane = tid & 31;
  const int wave = tid >> 5;
  const int ln   = lane & 15;
  const int hf   = lane >> 4;

  const int tileBase = (blockIdx.x * 8 + wave) * 16;

  // A-matrix row owned by this lane (row == edge within tile); clamp for tail waves
  const int aEdge = min(tileBase + ln, NEDGES - 1);
  const int aSrc  = srcI[aEdge];
  const int aTgt  = tgtI[aEdge];

  v8f acc[8];
#pragma unroll
  for (int nb = 0; nb < 8; ++nb) acc[nb] = (v8f)0.0f;

  const int kl0 = hf * 2;  // K sub-offset for this half-wave (A and B layouts)

  for (int kc = 0; kc < KE / 64; ++kc) {
    // cooperatively stage We[kc*64 .. +64)[0..128) transposed into LDS
#pragma unroll
    for (int i = 0; i < 32; ++i) {
      int idx = tid + i * 256;
      int kr  = idx >> 7;    // 0..63
      int col = idx & 127;
      sB[col * SB_STRIDE + kr] = We[(size_t)(kc * 64 + kr) * D + col];
    }
    __syncthreads();

    const float* aBase;
    if (kc < 2)      aBase = edgeF + (size_t)aEdge * D + kc * 64;
    else if (kc < 4) aBase = nodeF + (size_t)aSrc * D + (kc - 2) * 64;
    else             aBase = nodeF + (size_t)aTgt * D + (kc - 4) * 64;

#pragma unroll
    for (int k4 = 0; k4 < 16; ++k4) {
      const int kl = k4 * 4 + kl0;
      v2f a = *(const v2f*)(aBase + kl);
#pragma unroll
      for (int nb = 0; nb < 8; ++nb) {
        v2f b = *(const v2f*)(&sB[(nb * 16 + ln) * SB_STRIDE + kl]);
        acc[nb] = __builtin_amdgcn_wmma_f32_16x16x4_f32(
            false, a, false, b, (short)0, acc[nb], false, false);
      }
    }
    __syncthreads();
  }

  // per-lane parameter fragments across the 8 n-blocks (col = nb*16 + ln)
  float beF[8], gaF[8], btF[8];
#pragma unroll
  for (int nb = 0; nb < 8; ++nb) {
    int c = nb * 16 + ln;
    beF[nb] = be[c]; gaF[nb] = gammaE[c]; btF[nb] = betaE[c];
  }

  // C/D layout: element v of acc[nb] at this lane is row (v + 8*hf), col nb*16+ln
#pragma unroll
  for (int v = 0; v < 8; ++v) {
    const int rowEdge = tileBase + v + 8 * hf;
    const bool valid  = rowEdge < NEDGES;
    const int  er     = valid ? rowEdge : (NEDGES - 1);

    float vals[8];
    float psum = 0.0f;
#pragma unroll
    for (int nb = 0; nb < 8; ++nb) {
      int c = nb * 16 + ln;
      float x = acc[nb][v] + beF[nb] + edgeF[(size_t)er * D + c];
      vals[nb] = x;
      psum += x;
    }
    float mean = half16_reduce(psum) * (1.0f / D);
    float pvar = 0.0f;
#pragma unroll
    for (int nb = 0; nb < 8; ++nb) {
      float dd = vals[nb] - mean;
      vals[nb] = dd;
      pvar += dd * dd;
    }
    float var  = half16_reduce(pvar) * (1.0f / D);
    float rstd = __frsqrt_rn(var + LN_EPS);

    if (valid) {
      const int s = srcI[er];
      const int t = tgtI[er];
#pragma unroll
      for (int nb = 0; nb < 8; ++nb) {
        int c = nb * 16 + ln;
        float y = vals[nb] * rstd * gaF[nb] + btF[nb];
        newE[(size_t)er * D + c] = y;
        atomicAdd(&eSum[(size_t)s * D + c], y);
        atomicAdd(&eSum[(size_t)t * D + c], y);
      }
      if (ln == 0) {
        atomicAdd(&eCnt[s], 1.0f);
        atomicAdd(&eCnt[t], 1.0f);
      }
    }
  }
}

// ---------------- Node update: new_x = LN(x + [x, mean_e] @ Wn + bn) ----------------
__global__ __launch_bounds__(256) void node_kernel(
    const float* __restrict__ nodeF, const float* __restrict__ Wn,
    const float* __restrict__ bn, const float* __restrict__ gammaN,
    const float* __restrict__ betaN, const float* __restrict__ eSum,
    const float* __restrict__ eCnt, float* __restrict__ newN)
{
  __shared__ float sB[D * SB_STRIDE];
  const int tid  = threadIdx.x;
  const int lane = tid & 31;
  const int wave = tid >> 5;
  const int ln   = lane & 15;
  const int hf   = lane >> 4;

  const int tileBase = (blockIdx.x * 8 + wave) * 16;
  const int aNode = min(tileBase + ln, NNODES - 1);
  const float invC = 1.0f / (eCnt[aNode] + CNT_EPS);

  v8f acc[8];
#pragma unroll
  for (int nb = 0; nb < 8; ++nb) acc[nb] = (v8f)0.0f;

  const int kl0 = hf * 2;

  for (int kc = 0; kc < KN / 64; ++kc) {
#pragma unroll
    for (int i = 0; i < 32; ++i) {
      int idx = tid + i * 256;
      int kr  = idx >> 7;
      int col = idx & 127;
      sB[col * SB_STRIDE + kr] = Wn[(size_t)(kc * 64 + kr) * D + col];
    }
    __syncthreads();

    const float* aBase = (kc < 2)
        ? nodeF + (size_t)aNode * D + kc * 64
        : eSum  + (size_t)aNode * D + (kc - 2) * 64;
    const float aScale = (kc < 2) ? 1.0f : invC;  // mean = sum / (count + eps)

#pragma unroll
    for (int k4 = 0; k4 < 16; ++k4) {
      const int kl = k4 * 4 + kl0;
      v2f a = *(const v2f*)(aBase + kl);
      a.x *= aScale; a.y *= aScale;
#pragma unroll
      for (int nb = 0; nb < 8; ++nb) {
        v2f b = *(const v2f*)(&sB[(nb * 16 + ln) * SB_STRIDE + kl]);
        acc[nb] = __builtin_amdgcn_wmma_f32_16x16x4_f32(
            false, a, false, b, (short)0, acc[nb], false, false);
      }
    }
    __syncthreads();
  }

  float bnF[8], gaF[8], btF[8];
#pragma unroll
  for (int nb = 0; nb < 8; ++nb) {
    int c = nb * 16 + ln;
    bnF[nb] = bn[c]; gaF[nb] = gammaN[c]; btF[nb] = betaN[c];
  }

#pragma unroll
  for (int v = 0; v < 8; ++v) {
    const int rowNode = tileBase + v + 8 * hf;
    const bool valid  = rowNode < NNODES;
    const int  nr     = valid ? rowNode : (NNODES - 1);

    float vals[8];
    float psum = 0.0f;
#pragma unroll
    for (int nb = 0; nb < 8; ++nb) {
      int c = nb * 16 + ln;
      float x = acc[nb][v] + bnF[nb] + nodeF[(size_t)nr * D + c];
      vals[nb] = x;
      psum += x;
    }
    float mean = half16_reduce(psum) * (1.0f / D);
    float pvar = 0.0f;
#pragma unroll
    for (int nb = 0; nb < 8; ++nb) {
      float dd = vals[nb] - mean;
      vals[nb] = dd;
      pvar += dd * dd;
    }
    float var  = half16_reduce(pvar) * (1.0f / D);
    float rstd = __frsqrt_rn(var + LN_EPS);

    if (valid) {
#pragma unroll
      for (int nb = 0; nb < 8; ++nb) {
        int c = nb * 16 + ln;
        newN[(size_t)nr * D + c] = vals[nb] * rstd * gaF[nb] + btF[nb];
      }
    }
  }
}

extern "C" void kernel_launch(void* const* d_in, const int* in_sizes, int n_in,
                              void* d_out, int out_size, void* d_ws, size_t ws_size,
                              hipStream_t stream) {
  const float* nodeF  = (const float*)d_in[0];
  const float* edgeF  = (const float*)d_in[1];
  const int*   ei     = (const int*)d_in[2];
  const float* We     = (const float*)d_in[3];
  const float* be     = (const float*)d_in[4];
  const float* gammaE = (const float*)d_in[5];
  const float* betaE  = (const float*)d_in[6];
  const float* Wn     = (const float*)d_in[7];
  const float* bn     = (const float*)d_in[8];
  const float* gammaN = (const float*)d_in[9];
  const float* betaN  = (const float*)d_in[10];

  float* out  = (float*)d_out;
  float* newN = out;                                 // (N_NODES, 128) first
  float* newE = out + (size_t)NNODES * D;            // (N_EDGES, 128) second

  float* eSum = (float*)d_ws;                        // (N_NODES, 128)
  float* eCnt = eSum + (size_t)NNODES * D;           // (N_NODES,)

  const int* srcI = ei;
  const int* tgtI = ei + NEDGES;

  // 1) zero the scatter accumulators (harness does not re-zero between replays)
  size_t zn = (size_t)NNODES * D + NNODES;
  zero_kernel<<<4096, 256, 0, stream>>>(eSum, zn);

  // 2) edge update + scatter:  62500 tiles of 16 edges, 8 waves/block
  int edgeBlocks = (NEDGES / 16 + 7) / 8;            // 7813
  edge_kernel<<<edgeBlocks, 256, 0, stream>>>(
      nodeF, edgeF, srcI, tgtI, We, be, gammaE, betaE, newE, eSum, eCnt);

  // 3) node update: 6250 tiles of 16 nodes, 8 waves/block
  int nodeBlocks = (NNODES / 16 + 7) / 8;            // 782
  node_kernel<<<nodeBlocks, 256, 0, stream>>>(
      nodeF, Wn, bn, gammaN, betaN, eSum, eCnt, newN);
}